// Network_12257836663218
// MI455X (gfx1250) — compile-verified
//
#include <hip/hip_runtime.h>
#include <math.h>

#define IMG 28
#define NU 96
#define SDIM 2688            // IMG*NU
#define NB 9216              // NU*NU
#define SS 7225344LL         // SDIM*SDIM
#define RV_ALPHA 0.9f
#define EPS_LOG 1e-08f

// ---------------- CDNA5 async global->LDS helpers ----------------

// Matches the builtin's parameter pointee type exactly (per hipcc diagnostic):
// 'int __attribute__((__vector_size__(4 * sizeof(int))))'
typedef int v4i_vs __attribute__((vector_size(16)));

__device__ __forceinline__ void async_copy_b128(void* lds_dst, const void* gsrc) {
#if defined(__AMDGCN__)
#if __has_builtin(__builtin_amdgcn_global_load_async_to_lds_b128)
  __builtin_amdgcn_global_load_async_to_lds_b128(
      (__attribute__((address_space(1))) v4i_vs*)(gsrc),
      (__attribute__((address_space(3))) v4i_vs*)(lds_dst),
      /*offset=*/0, /*cpol=*/0);
#else
  unsigned lds_off =
      (unsigned)(__SIZE_TYPE__)(__attribute__((address_space(3))) char*)(lds_dst);
  unsigned long long ga = (unsigned long long)gsrc;
  asm volatile("global_load_async_to_lds_b128 %0, %1, off"
               :: "v"(lds_off), "v"(ga) : "memory");
#endif
#else
  (void)lds_dst; (void)gsrc;
#endif
}

__device__ __forceinline__ void wait_async_le2() {
#if defined(__AMDGCN__)
  asm volatile("s_wait_asynccnt 2" ::: "memory");
#endif
}

__device__ __forceinline__ void wait_async_le0() {
#if defined(__AMDGCN__)
  asm volatile("s_wait_asynccnt 0" ::: "memory");
#endif
}

// ---------------- Kernel 1: per-unit tile reduction -> unit_map ----------------
// One workgroup per (u,v) unit: sum over its 28x28 tile of (som - x)^2 / rv.

__global__ __launch_bounds__(128) void k_unit_map(
    const float* __restrict__ som, const float* __restrict__ rv,
    const float* __restrict__ x, float* __restrict__ umap) {
  int b = blockIdx.x;                 // 0..NB-1
  int u = b / NU, v = b - u * NU;
  long long base = (long long)(u * IMG) * SDIM + (long long)(v * IMG);
  float acc = 0.0f;
  for (int a = threadIdx.x; a < IMG * IMG; a += 128) {
    int r = a / IMG, c = a - r * IMG;
    long long g = base + (long long)r * SDIM + c;
    float t = x[a];                   // a == r*IMG + c
    float d = som[g] - t;
    acc += d * d / rv[g];
  }
  __shared__ float red[128];
  red[threadIdx.x] = acc;
  __syncthreads();
  for (int s = 64; s > 0; s >>= 1) {
    if (threadIdx.x < (unsigned)s) red[threadIdx.x] += red[threadIdx.x + s];
    __syncthreads();
  }
  if (threadIdx.x == 0) umap[b] = red[0];
}

// ---------------- Kernel 2: argmin + BMU modifier maps ----------------

__global__ __launch_bounds__(256) void k_bmu_mods(
    const float* __restrict__ umap, const float* __restrict__ lrates,
    const float* __restrict__ radius, const float* __restrict__ cart,
    float* __restrict__ fm, float* __restrict__ va) {
  __shared__ float svals[256];
  __shared__ int   sidx[256];
  int tid = threadIdx.x;

  float best = INFINITY; int bidx = 0x7fffffff;
  for (int i = tid; i < NB; i += 256) {
    float vv = umap[i];
    if (vv < best || (vv == best && i < bidx)) { best = vv; bidx = i; }
  }
  svals[tid] = best; sidx[tid] = bidx;
  __syncthreads();
  for (int s = 128; s > 0; s >>= 1) {
    if (tid < s) {
      float ov = svals[tid + s]; int oi = sidx[tid + s];
      if (ov < svals[tid] || (ov == svals[tid] && oi < sidx[tid])) {
        svals[tid] = ov; sidx[tid] = oi;
      }
    }
    __syncthreads();
  }
  int flat = sidx[0];                     // bi*NU + bj, first-occurrence argmin
  float r  = radius[flat];
  float lr = lrates[flat];
  float dist_mod = 1.0f / (2.0f * r * r);
  float constant = -logf(EPS_LOG / lr) / dist_mod;

  for (int i = tid; i < NB; i += 256) {
    // cartesian_distances shape (NU,NU,NU,NU): [iu,jv,bi,bj] = i*NB + flat
    float d = cart[(long long)i * NB + flat];
    float mask = (d > r) ? 0.0f : 1.0f;
    float fmod = mask * lrates[i] * __expf(-d * dist_mod);
    float valpha = RV_ALPHA - 0.5f + 1.0f / (1.0f + __expf(-d / constant));
    valpha = valpha * mask + (1.0f - mask);
    valpha = fminf(fmaxf(valpha, 0.0f), 1.0f);
    fm[i] = fmod;
    va[i] = valpha;
  }
}

// ---------------- Kernel 3: streaming update with async double-buffering ----------------
// grid = 1008 blocks * 256 threads * 7 iters * float4 == SDIM*SDIM exactly.

#define K3_BLOCKS 1008
#define K3_ITERS  7

__global__ __launch_bounds__(256) void k_update(
    const float4* __restrict__ som4, const float4* __restrict__ rv4,
    const float* __restrict__ x, const float* __restrict__ fm,
    const float* __restrict__ va, float* __restrict__ out) {
  __shared__ float  xs[IMG * IMG];
  __shared__ float4 sbuf[2][256];
  __shared__ float4 rbuf[2][256];
  int tid = threadIdx.x;

  // Stage x (3.1 KB, reused by every element) into LDS once.
  for (int t = tid; t < IMG * IMG; t += 256) xs[t] = x[t];
  __syncthreads();

  const long long stride4 = (long long)K3_BLOCKS * 256;
  long long lane0 = (long long)blockIdx.x * 256 + tid;

  // Prefetch chunk 0 (async DMA into this lane's LDS slots).
  async_copy_b128(&sbuf[0][tid], som4 + lane0);
  async_copy_b128(&rbuf[0][tid], rv4 + lane0);

  for (int k = 0; k < K3_ITERS; ++k) {
    long long cur = lane0 + (long long)k * stride4;     // float4 index
    if (k + 1 < K3_ITERS) {
      long long nxt = cur + stride4;
      async_copy_b128(&sbuf[(k + 1) & 1][tid], som4 + nxt);
      async_copy_b128(&rbuf[(k + 1) & 1][tid], rv4 + nxt);
      wait_async_le2();   // chunk k (2 oldest ops) complete; next 2 in flight
    } else {
      wait_async_le0();   // drain
    }

    float4 sv = sbuf[k & 1][tid];
    float4 rq = rbuf[k & 1][tid];
    float so[4] = {sv.x, sv.y, sv.z, sv.w};
    float rr[4] = {rq.x, rq.y, rq.z, rq.w};
    float o0[4], o1[4];

    long long ebase = cur * 4;
#pragma unroll
    for (int c = 0; c < 4; ++c) {
      long long idx = ebase + c;
      int i = (int)(idx / SDIM);
      int j = (int)(idx - (long long)i * SDIM);
      int u  = i / IMG;
      int vv = j / IMG;
      int xi = i - u * IMG;
      int xj = j - vv * IMG;
      float t = xs[xi * IMG + xj];
      float f = fm[u * NU + vv];
      float a = va[u * NU + vv];
      float s = so[c];
      float sn = s + f * (t - s);
      sn = fminf(fmaxf(sn, 0.0f), 1.0f);
      float dts = t - sn;
      float vn = a * rr[c] + (1.0f - a) * dts * dts;
      o0[c] = sn;
      o1[c] = vn;
    }
    float4 os = make_float4(o0[0], o0[1], o0[2], o0[3]);
    float4 ov = make_float4(o1[0], o1[1], o1[2], o1[3]);
    ((float4*)out)[cur] = os;
    ((float4*)(out + SS))[cur] = ov;
  }
}

// ---------------- Host launcher ----------------

extern "C" void kernel_launch(void* const* d_in, const int* in_sizes, int n_in,
                              void* d_out, int out_size, void* d_ws, size_t ws_size,
                              hipStream_t stream) {
  const float* som  = (const float*)d_in[0];
  const float* rv   = (const float*)d_in[1];
  const float* lr   = (const float*)d_in[2];
  const float* rad  = (const float*)d_in[3];
  const float* cart = (const float*)d_in[4];
  const float* x    = (const float*)d_in[5];
  float* out = (float*)d_out;

  float* umap = (float*)d_ws;      // NB floats
  float* fm   = umap + NB;         // NB floats
  float* va   = fm + NB;           // NB floats  (total 108 KB of d_ws)

  k_unit_map<<<NB, 128, 0, stream>>>(som, rv, x, umap);
  k_bmu_mods<<<1, 256, 0, stream>>>(umap, lr, rad, cart, fm, va);
  k_update<<<K3_BLOCKS, 256, 0, stream>>>((const float4*)som, (const float4*)rv,
                                          x, fm, va, out);
}